// SparXSS2D_18391049962075
// MI455X (gfx1250) — compile-verified
//
#include <hip/hip_runtime.h>

// ---------------------------------------------------------------------------
// SS2D (VMamba selective-scan 2D) for gfx1250.
// B=2, H=W=56, L=3136, d_model=d_inner=96, K=4 dirs, d_state=16, dt_rank=6.
// GEMMs run on V_WMMA_F32_16X16X4_F32 (f32 matrix pipe, exact f32 accumulate).
// The selective scan parallelizes over (channel, state): one lane per state,
// wave32 shuffle reduction (width 16) for y = sum_n h*C.
// ---------------------------------------------------------------------------

constexpr int D_M    = 96;            // d_model == d_inner
constexpr int NSTATE = 16;
constexpr int RANK   = 6;
constexpr int KD     = 4;
constexpr int NB     = 2;
constexpr int H_     = 56;
constexpr int W_     = 56;
constexpr int L_     = H_ * W_;       // 3136 = 49*64
constexpr int EDIM   = RANK + 2 * NSTATE;  // 38

typedef __attribute__((ext_vector_type(2))) float v2f;
typedef __attribute__((ext_vector_type(8))) float v8f;

// direction remap: column l of direction-k sequence -> spatial index into (H,W)
__device__ __forceinline__ int dir_map(int k, int l) {
  switch (k & 3) {
    case 0:  return l;
    case 1:  return (l % H_) * W_ + (l / H_);
    case 2:  return L_ - 1 - l;
    default: { int l2 = L_ - 1 - l; return (l2 % H_) * W_ + (l2 / H_); }
  }
}

// ---------------------------------------------------------------------------
// Y[b] = Wm(96x96) @ X[b](96 x L) + bias        (in_proj and out_proj)
// One wave computes a 16(M) x 64(N) tile; K=96 in 24 steps of WMMA f32 K=4.
// A frag (16x4): lanes 0-15 row m, {K0,K1}; lanes 16-31 row m, {K2,K3}.
// ---------------------------------------------------------------------------
__global__ __launch_bounds__(32)
void gemm96_kernel(const float* __restrict__ Wm, const float* __restrict__ bias,
                   const float* __restrict__ X, float* __restrict__ Y)
{
  const int lane = threadIdx.x & 31;
  const int lh   = lane & 15;
  const bool hi  = lane >= 16;
  const int n0   = blockIdx.x * 64;
  const int m0   = blockIdx.y * 16;
  const int b    = blockIdx.z;
  const float* Xb = X + (size_t)b * D_M * L_;
  float*       Yb = Y + (size_t)b * D_M * L_;

  v8f acc0 = {}, acc1 = {}, acc2 = {}, acc3 = {};
  const int n = n0 + lh;
  for (int kk = 0; kk < 24; ++kk) {
    const int k0 = kk * 4 + (hi ? 2 : 0);
    v2f a;
    a.x = Wm[(m0 + lh) * D_M + k0];
    a.y = Wm[(m0 + lh) * D_M + k0 + 1];
    const float* x0 = Xb + (size_t)k0 * L_;
    const float* x1 = x0 + L_;
    v2f b0, b1, b2, b3;
    b0.x = x0[n];      b0.y = x1[n];
    b1.x = x0[n + 16]; b1.y = x1[n + 16];
    b2.x = x0[n + 32]; b2.y = x1[n + 32];
    b3.x = x0[n + 48]; b3.y = x1[n + 48];
    acc0 = __builtin_amdgcn_wmma_f32_16x16x4_f32(false, a, false, b0, (short)0, acc0, false, false);
    acc1 = __builtin_amdgcn_wmma_f32_16x16x4_f32(false, a, false, b1, (short)0, acc1, false, false);
    acc2 = __builtin_amdgcn_wmma_f32_16x16x4_f32(false, a, false, b2, (short)0, acc2, false, false);
    acc3 = __builtin_amdgcn_wmma_f32_16x16x4_f32(false, a, false, b3, (short)0, acc3, false, false);
  }
  const int mb = m0 + (hi ? 8 : 0);
  #pragma unroll
  for (int r = 0; r < 8; ++r) {
    const int m = mb + r;
    const float bv = bias[m];
    float* yr = Yb + (size_t)m * L_ + n0 + lh;
    yr[0]  = acc0[r] + bv;
    yr[16] = acc1[r] + bv;
    yr[32] = acc2[r] + bv;
    yr[48] = acc3[r] + bv;
  }
}

// ---------------------------------------------------------------------------
// x_dbl[b,k,e,l] = sum_d xw[k,e,d] * hs[b,d,dir_map(k,l)]   (cross_scan fused)
// M = 38 padded to 48 (3 tiles); rows >= 38 use zero weights, never stored.
// ---------------------------------------------------------------------------
__global__ __launch_bounds__(32)
void xproj_kernel(const float* __restrict__ XW,   // (K,38,96)
                  const float* __restrict__ HS,   // (B,96,L)
                  float* __restrict__ XD)         // (B,K,38,L)
{
  const int lane = threadIdx.x & 31;
  const int lh   = lane & 15;
  const bool hi  = lane >= 16;
  const int n0   = blockIdx.x * 64;
  const int m0   = blockIdx.y * 16;
  const int bk   = blockIdx.z;            // b*K + k
  const int k    = bk % KD;
  const int b    = bk / KD;
  const float* Hb = HS + (size_t)b * D_M * L_;
  const float* Wk = XW + (size_t)k * EDIM * D_M;
  float*       Ob = XD + (size_t)bk * EDIM * L_;

  const int cs0 = dir_map(k, n0 + lh);
  const int cs1 = dir_map(k, n0 + 16 + lh);
  const int cs2 = dir_map(k, n0 + 32 + lh);
  const int cs3 = dir_map(k, n0 + 48 + lh);

  const int  e  = m0 + lh;
  const bool ev = e < EDIM;

  v8f acc0 = {}, acc1 = {}, acc2 = {}, acc3 = {};
  for (int kk = 0; kk < 24; ++kk) {
    const int k0 = kk * 4 + (hi ? 2 : 0);
    v2f a;
    a.x = ev ? Wk[e * D_M + k0]     : 0.0f;
    a.y = ev ? Wk[e * D_M + k0 + 1] : 0.0f;
    const float* h0 = Hb + (size_t)k0 * L_;
    const float* h1 = h0 + L_;
    v2f b0, b1, b2, b3;
    b0.x = h0[cs0]; b0.y = h1[cs0];
    b1.x = h0[cs1]; b1.y = h1[cs1];
    b2.x = h0[cs2]; b2.y = h1[cs2];
    b3.x = h0[cs3]; b3.y = h1[cs3];
    acc0 = __builtin_amdgcn_wmma_f32_16x16x4_f32(false, a, false, b0, (short)0, acc0, false, false);
    acc1 = __builtin_amdgcn_wmma_f32_16x16x4_f32(false, a, false, b1, (short)0, acc1, false, false);
    acc2 = __builtin_amdgcn_wmma_f32_16x16x4_f32(false, a, false, b2, (short)0, acc2, false, false);
    acc3 = __builtin_amdgcn_wmma_f32_16x16x4_f32(false, a, false, b3, (short)0, acc3, false, false);
  }
  const int mb = m0 + (hi ? 8 : 0);
  #pragma unroll
  for (int r = 0; r < 8; ++r) {
    const int m = mb + r;
    if (m < EDIM) {
      float* orow = Ob + (size_t)m * L_ + n0 + lh;
      orow[0]  = acc0[r];
      orow[16] = acc1[r];
      orow[32] = acc2[r];
      orow[48] = acc3[r];
    }
  }
}

// ---------------------------------------------------------------------------
// depthwise 3x3 conv + bias + SiLU, zero padding
// ---------------------------------------------------------------------------
__global__ void dwconv_silu_kernel(const float* __restrict__ Hin,
                                   const float* __restrict__ CW,  // (96,1,3,3)
                                   const float* __restrict__ CB,
                                   float* __restrict__ Hout)
{
  const int idx = blockIdx.x * blockDim.x + threadIdx.x;
  if (idx >= NB * D_M * L_) return;
  const int w  = idx % W_;
  const int hh = (idx / W_) % H_;
  const int d  = (idx / L_) % D_M;
  const float* src = Hin + (size_t)(idx / L_) * L_;   // (b*96+d) plane
  float acc = CB[d];
  #pragma unroll
  for (int ky = 0; ky < 3; ++ky) {
    const int yy = hh + ky - 1;
    if (yy < 0 || yy >= H_) continue;
    #pragma unroll
    for (int kx = 0; kx < 3; ++kx) {
      const int xx = w + kx - 1;
      if (xx < 0 || xx >= W_) continue;
      acc += CW[d * 9 + ky * 3 + kx] * src[yy * W_ + xx];
    }
  }
  Hout[idx] = acc / (1.0f + __expf(-acc));   // SiLU
}

// ---------------------------------------------------------------------------
// delta[b,k,d,l] = softplus( dtw[k,d,:] . x_dbl[b,k,0:6,l] + dtb[k,d] )
// ---------------------------------------------------------------------------
__global__ void delta_kernel(const float* __restrict__ XD,   // (B,K,38,L)
                             const float* __restrict__ DTW,  // (K,96,6)
                             const float* __restrict__ DTB,  // (K,96)
                             float* __restrict__ DEL)        // (B,K,96,L)
{
  const int idx = blockIdx.x * blockDim.x + threadIdx.x;
  if (idx >= NB * KD * D_M * L_) return;
  const int l = idx % L_;
  const int d = (idx / L_) % D_M;
  const int k = (idx / (L_ * D_M)) % KD;
  const int b = idx / (L_ * D_M * KD);
  const float* xr = XD + (size_t)(b * KD + k) * EDIM * L_ + l;
  const float* wr = DTW + (size_t)(k * D_M + d) * RANK;
  float t = DTB[k * D_M + d];
  #pragma unroll
  for (int r = 0; r < RANK; ++r) t += wr[r] * xr[(size_t)r * L_];
  DEL[idx] = (t > 20.0f) ? t : log1pf(__expf(t));   // softplus
}

// ---------------------------------------------------------------------------
// selective scan: one lane per (channel, state). 16-lane shuffle reduction
// for y[l] = sum_n h[l,n]*C[n,l]; lane n==0 adds D*u and stores.
// ---------------------------------------------------------------------------
__global__ __launch_bounds__(256)
void scan_kernel(const float* __restrict__ DEL,   // (B,K,96,L)
                 const float* __restrict__ HS,    // (B,96,L)   u via dir_map
                 const float* __restrict__ XD,    // (B,K,38,L) Bs rows 6..21, Cs 22..37
                 const float* __restrict__ ALOG,  // (384,16)
                 const float* __restrict__ DS,    // (384,)
                 float* __restrict__ Yo)          // (B,K,96,L)
{
  const int n  = threadIdx.x & 15;                 // state index
  const int g  = threadIdx.x >> 4;                 // group in block
  const int ch = blockIdx.x * 16 + g;              // 0..767 == (b*K+k)*96+d
  const int b  = ch / (KD * D_M);
  const int k  = (ch / D_M) % KD;
  const int d  = ch % D_M;

  const float an  = -__expf(ALOG[(size_t)(k * D_M + d) * NSTATE + n]);
  const float dsv = DS[k * D_M + d];

  const float* dl = DEL + (size_t)ch * L_;
  const float* up = HS  + (size_t)(b * D_M + d) * L_;
  const float* Bp = XD  + ((size_t)(b * KD + k) * EDIM + RANK + n) * L_;
  const float* Cp = Bp + (size_t)NSTATE * L_;
  float*       yp = Yo + (size_t)ch * L_;

  float hstate = 0.0f;
  for (int l = 0; l < L_; ++l) {
    const float de = dl[l];
    const float ul = up[dir_map(k, l)];
    const float dA = __expf(de * an);
    hstate = dA * hstate + (de * ul) * Bp[l];
    float yq = hstate * Cp[l];
    yq += __shfl_xor(yq, 1, 16);
    yq += __shfl_xor(yq, 2, 16);
    yq += __shfl_xor(yq, 4, 16);
    yq += __shfl_xor(yq, 8, 16);
    if (n == 0) yp[l] = yq + dsv * ul;
  }
}

// ---------------------------------------------------------------------------
// cross_merge (4-direction gather-sum) + LayerNorm over the 96 channels.
// One wave per (b,l): lane covers channels lane, lane+32, lane+64.
// ---------------------------------------------------------------------------
__global__ __launch_bounds__(256)
void merge_ln_kernel(const float* __restrict__ Yo,     // (B,K,96,L)
                     const float* __restrict__ gamma,
                     const float* __restrict__ beta,
                     float* __restrict__ Z)             // (B,96,L)
{
  const int wid  = blockIdx.x * 8 + (threadIdx.x >> 5);
  const int lane = threadIdx.x & 31;
  const int b  = wid / L_;
  const int l  = wid % L_;
  const int hh = l / W_;
  const int w  = l % W_;
  const int lt = w * H_ + hh;                     // transposed position
  const size_t base = (size_t)b * KD * D_M * L_;

  float v[3];
  float s = 0.f, s2 = 0.f;
  #pragma unroll
  for (int i = 0; i < 3; ++i) {
    const int d = lane + 32 * i;
    float t = Yo[base + (size_t)(0 * D_M + d) * L_ + l]
            + Yo[base + (size_t)(1 * D_M + d) * L_ + lt]
            + Yo[base + (size_t)(2 * D_M + d) * L_ + (L_ - 1 - l)]
            + Yo[base + (size_t)(3 * D_M + d) * L_ + (L_ - 1 - lt)];
    v[i] = t; s += t; s2 += t * t;
  }
  #pragma unroll
  for (int m = 1; m < 32; m <<= 1) {
    s  += __shfl_xor(s,  m, 32);
    s2 += __shfl_xor(s2, m, 32);
  }
  const float mean = s * (1.0f / 96.0f);
  const float var  = s2 * (1.0f / 96.0f) - mean * mean;
  const float inv  = rsqrtf(var + 1e-5f);
  #pragma unroll
  for (int i = 0; i < 3; ++i) {
    const int d = lane + 32 * i;
    Z[((size_t)b * D_M + d) * L_ + l] = (v[i] - mean) * inv * gamma[d] + beta[d];
  }
}

// ---------------------------------------------------------------------------
extern "C" void kernel_launch(void* const* d_in, const int* in_sizes, int n_in,
                              void* d_out, int out_size, void* d_ws, size_t ws_size,
                              hipStream_t stream)
{
  const float* x    = (const float*)d_in[0];
  const float* ipw  = (const float*)d_in[1];
  const float* ipb  = (const float*)d_in[2];
  const float* cw   = (const float*)d_in[3];
  const float* cb   = (const float*)d_in[4];
  const float* xpw  = (const float*)d_in[5];
  const float* dtw  = (const float*)d_in[6];
  const float* dtb  = (const float*)d_in[7];
  const float* alog = (const float*)d_in[8];
  const float* ds   = (const float*)d_in[9];
  const float* gam  = (const float*)d_in[10];
  const float* bet  = (const float*)d_in[11];
  const float* opw  = (const float*)d_in[12];
  const float* opb  = (const float*)d_in[13];
  float* out = (float*)d_out;

  float* ws = (float*)d_ws;
  const size_t szBCL  = (size_t)NB * D_M * L_;          // 602112
  const size_t szXD   = (size_t)NB * KD * EDIM * L_;    // 953344
  const size_t szBKDL = (size_t)NB * KD * D_M * L_;     // 2408448
  float* h   = ws;  ws += szBCL;
  float* hs  = ws;  ws += szBCL;
  float* xd  = ws;  ws += szXD;
  float* del = ws;  ws += szBKDL;
  float* yo  = ws;  ws += szBKDL;
  float* z   = ws;  ws += szBCL;

  // 1) in_proj GEMM (WMMA f32)
  gemm96_kernel<<<dim3(L_ / 64, D_M / 16, NB), 32, 0, stream>>>(ipw, ipb, x, h);
  // 2) depthwise conv + SiLU
  dwconv_silu_kernel<<<(NB * D_M * L_ + 255) / 256, 256, 0, stream>>>(h, cw, cb, hs);
  // 3) x_proj GEMM with fused cross_scan gather (WMMA f32)
  xproj_kernel<<<dim3(L_ / 64, 3, NB * KD), 32, 0, stream>>>(xpw, hs, xd);
  // 4) dt projection + softplus
  delta_kernel<<<(NB * KD * D_M * L_ + 255) / 256, 256, 0, stream>>>(xd, dtw, dtb, del);
  // 5) selective scan (lane-per-state, shuffle reduce)
  scan_kernel<<<(NB * KD * D_M) / 16, 256, 0, stream>>>(del, hs, xd, alog, ds, yo);
  // 6) cross_merge + LayerNorm
  merge_ln_kernel<<<(NB * L_) / 8, 256, 0, stream>>>(yo, gam, bet, z);
  // 7) out_proj GEMM (WMMA f32)
  gemm96_kernel<<<dim3(L_ / 64, D_M / 16, NB), 32, 0, stream>>>(opw, opb, z, out);
}